// LaplacianODEFunc_Polynomial_9174050144893
// MI455X (gfx1250) — compile-verified
//
#include <hip/hip_runtime.h>
#include <stdint.h>

// LaplacianODEFunc Chebyshev polynomial filter for MI455X (gfx1250).
//
// Roofline: 12.3 GFLOP vs ~25 GB of gather traffic. One dense (N,128) f32
// matrix = 102.4 MB < 192 MB L2, so gathers/atomic scatters are L2-resident;
// HBM sees mostly the 15x edge stream (~1 GB) -> bandwidth/atomic bound,
// WMMA inapplicable (sparse density 16/200000). CDNA5 paths used instead:
// wave32 edge-parallel float4 gathers (global_load_b128), native f32 global
// atomics, and Tensor Data Mover DMA (tensor_load_to_lds + s_wait_tensorcnt)
// double-buffering the COO edge stream into LDS.

#define H_CH   128   // hidden channels (reference H)
#define TILE   512   // edges staged per block tile (TDM DMA granularity)
#define NWAVES 8     // 256 threads / wave32

typedef __attribute__((ext_vector_type(4))) unsigned int v4u;
typedef __attribute__((ext_vector_type(8))) unsigned int v8u;

// ---------------- tiny device softmax over poly logits ----------------
__global__ void k_softmax(const float* __restrict__ logits, int n,
                          float* __restrict__ w) {
  if (threadIdx.x == 0 && blockIdx.x == 0) {
    float m = logits[0];
    for (int i = 1; i < n; ++i) m = fmaxf(m, logits[i]);
    float s = 0.f;
    for (int i = 0; i < n; ++i) s += __expf(logits[i] - m);
    float inv = 1.f / s;
    for (int i = 0; i < n; ++i) w[i] = __expf(logits[i] - m) * inv;
  }
}

// ---------------- zero a float4 buffer (SpMM accumulator) ----------------
__global__ void k_zero(float4* __restrict__ p, long n4) {
  long i = (long)blockIdx.x * blockDim.x + threadIdx.x;
  const long stride = (long)gridDim.x * blockDim.x;
  const float4 z = {0.f, 0.f, 0.f, 0.f};
  for (; i < n4; i += stride) p[i] = z;
}

// ---------------- TDM: 1-D tile DMA, global -> LDS ----------------
// Builds the Tensor DMA Descriptor (D#) per CDNA5 ISA ch.8:
//  Group0: count=1 | lds_addr | global_addr | type=2 ("image")
//  Group1: data_size=4B, tensor_dim0=remaining (hw zero-fills OOB),
//          tensor_dim1=1, tile_dim0=nelem, tile_dim1..2=0 (1-D), strides=0
//  Groups 2/3: zero (dims 2..4 unused)
// Issued once per wave (TDM ignores EXEC); caller gates to wave 0.
__device__ __forceinline__ void tdm_load_1d(uint32_t lds_off, const void* g,
                                            uint32_t avail_elems,
                                            uint32_t tile_elems) {
  const uint64_t ga = (uint64_t)(uintptr_t)g;
  v4u g0;
  g0.x = 1u;                                     // count=1, user descriptor
  g0.y = lds_off;                                // lds_addr (bytes)
  g0.z = (uint32_t)ga;                           // global_addr[31:0]
  g0.w = (uint32_t)(ga >> 32) | (2u << 30);      // global_addr[56:32] | type=2
  v8u g1;
  g1.s0 = 2u << 16;                              // data_size = 4 bytes
  g1.s1 = (avail_elems & 0xFFFFu) << 16;         // tensor_dim0[15:0]
  g1.s2 = ((avail_elems >> 16) & 0xFFFFu)        // tensor_dim0[31:16]
        | (1u << 16);                            // tensor_dim1 = 1
  g1.s3 = (tile_elems & 0xFFFFu) << 16;          // tile_dim0
  g1.s4 = 0u;                                    // tile_dim1=0, tile_dim2=0
  g1.s5 = 0u;                                    // tensor_dim0_stride (unused)
  g1.s6 = 0u;
  g1.s7 = 0u;                                    // tensor_dim1_stride (unused)
  v4u gz = {0u, 0u, 0u, 0u};                     // groups 2/3: dims 2..4 unused
  asm volatile("tensor_load_to_lds %0, %1, %2, %3"
               :: "s"(g0), "s"(g1), "s"(gz), "s"(gz)
               : "memory");
}

// ---------------- COO SpMM: dst += L @ src (dst pre-zeroed) ----------------
// One wave per edge: 32 lanes x float4 = 128 channels, fully coalesced.
// Edge metadata (val,row,col) double-buffered in LDS: wave 0 issues 3 TDM
// DMAs per tile for the NEXT tile, waits TENSORcnt<=3 (in-order completion
// => previous tile's DMAs finished), then all 8 waves barrier and process.
__launch_bounds__(256, 2)
__global__ void k_spmm(const float* __restrict__ vals,
                       const int*   __restrict__ erow,
                       const int*   __restrict__ ecol,
                       const float* __restrict__ src,
                       float*       __restrict__ dst,
                       long nnz) {
  __shared__ float sval[2][TILE];
  __shared__ int   srow[2][TILE];
  __shared__ int   scol[2][TILE];

  const int  tid    = threadIdx.x;
  const int  lane   = tid & 31;
  const int  wave   = tid >> 5;
  const long ntiles = (nnz + TILE - 1) / TILE;

  long t = blockIdx.x;
  if (t >= ntiles) return;

  if (wave == 0) {  // prologue: DMA tile t into buffer 0
    const long base = t * TILE;
    const uint32_t avail = (uint32_t)(nnz - base);
    tdm_load_1d((uint32_t)(uintptr_t)&sval[0][0], vals + base, avail, TILE);
    tdm_load_1d((uint32_t)(uintptr_t)&srow[0][0], erow + base, avail, TILE);
    tdm_load_1d((uint32_t)(uintptr_t)&scol[0][0], ecol + base, avail, TILE);
  }

  int cur = 0;
  for (; t < ntiles; t += gridDim.x) {
    const long nt = t + gridDim.x;
    if (wave == 0) {
      if (nt < ntiles) {
        const long nb = nt * TILE;
        const uint32_t avail = (uint32_t)(nnz - nb);
        tdm_load_1d((uint32_t)(uintptr_t)&sval[cur ^ 1][0], vals + nb, avail, TILE);
        tdm_load_1d((uint32_t)(uintptr_t)&srow[cur ^ 1][0], erow + nb, avail, TILE);
        tdm_load_1d((uint32_t)(uintptr_t)&scol[cur ^ 1][0], ecol + nb, avail, TILE);
        // 3 newest DMAs may remain in flight; previous tile's 3 are done
        __builtin_amdgcn_s_wait_tensorcnt((short)3);
      } else {
        __builtin_amdgcn_s_wait_tensorcnt((short)0);
      }
    }
    __syncthreads();

    // prefetch edge stream two tiles ahead into GL2
    {
      long pf = (t + 2 * (long)gridDim.x) * TILE + tid;
      if (pf < nnz) {
        __builtin_prefetch(vals + pf, 0, 1);
        __builtin_prefetch(erow + pf, 0, 1);
        __builtin_prefetch(ecol + pf, 0, 1);
      }
    }

    const long base = t * TILE;
    const long rem  = nnz - base;
    const int  lim  = (int)((rem < TILE) ? rem : TILE);
    for (int i = wave; i < lim; i += NWAVES) {
      const float v = sval[cur][i];
      const int   r = srow[cur][i];
      const int   c = scol[cur][i];
      const float4 g = ((const float4*)(src + (size_t)c * H_CH))[lane];
      float* o = dst + (size_t)r * H_CH + (size_t)lane * 4;
      unsafeAtomicAdd(o + 0, v * g.x);   // global_atomic_add_f32 (no return)
      unsafeAtomicAdd(o + 1, v * g.y);
      unsafeAtomicAdd(o + 2, v * g.z);
      unsafeAtomicAdd(o + 3, v * g.w);
    }
    __syncthreads();   // buffer 'cur' becomes a DMA target 2 tiles later
    cur ^= 1;
  }
}

// ---------------- fused first step ----------------
// T1  = S0 - x                         (L_hat x, since 2/lambda_max == 1)
// acc = alpha*(x - 0.5*S0) - (w0*x + w1*T1)   (hp term + sign folded in)
__global__ void k_init(const float4* __restrict__ x,
                       const float4* __restrict__ S0,
                       const float*  __restrict__ w,
                       const float*  __restrict__ alpha,
                       float4* __restrict__ T1,
                       float4* __restrict__ acc, long n4) {
  const float w0 = w[0], w1 = w[1], a = alpha[0];
  long i = (long)blockIdx.x * blockDim.x + threadIdx.x;
  const long stride = (long)gridDim.x * blockDim.x;
  for (; i < n4; i += stride) {
    const float4 xv = x[i];
    const float4 s  = S0[i];
    float4 t1, o;
    t1.x = s.x - xv.x; t1.y = s.y - xv.y; t1.z = s.z - xv.z; t1.w = s.w - xv.w;
    o.x = a * (xv.x - 0.5f * s.x) - (w0 * xv.x + w1 * t1.x);
    o.y = a * (xv.y - 0.5f * s.y) - (w0 * xv.y + w1 * t1.y);
    o.z = a * (xv.z - 0.5f * s.z) - (w0 * xv.z + w1 * t1.z);
    o.w = a * (xv.w - 0.5f * s.w) - (w0 * xv.w + w1 * t1.w);
    T1[i]  = t1;
    acc[i] = o;
  }
}

// ---------------- Chebyshev step ----------------
// Tn = 2*(S - T1) - T0 ; acc -= w[k]*Tn.  Tn may alias T0 (per-element RMW),
// so no __restrict__ on those two.
__global__ void k_step(const float4* __restrict__ S,
                       const float4* __restrict__ T1,
                       const float4* T0,
                       float4*       Tn,
                       float4* __restrict__ acc,
                       const float* __restrict__ w, int k, long n4) {
  const float wk = w[k];
  long i = (long)blockIdx.x * blockDim.x + threadIdx.x;
  const long stride = (long)gridDim.x * blockDim.x;
  for (; i < n4; i += stride) {
    const float4 s  = S[i];
    const float4 t1 = T1[i];
    const float4 t0 = T0[i];
    float4 tn;
    tn.x = 2.f * (s.x - t1.x) - t0.x;
    tn.y = 2.f * (s.y - t1.y) - t0.y;
    tn.z = 2.f * (s.z - t1.z) - t0.z;
    tn.w = 2.f * (s.w - t1.w) - t0.w;
    Tn[i] = tn;
    float4 av = acc[i];
    av.x -= wk * tn.x; av.y -= wk * tn.y;
    av.z -= wk * tn.z; av.w -= wk * tn.w;
    acc[i] = av;
  }
}

extern "C" void kernel_launch(void* const* d_in, const int* in_sizes, int n_in,
                              void* d_out, int out_size, void* d_ws,
                              size_t ws_size, hipStream_t stream) {
  const float* x      = (const float*)d_in[0];   // (N, 128) f32
  const float* vals   = (const float*)d_in[1];   // (NNZ,)  f32
  const float* logits = (const float*)d_in[2];   // (K+1,)  f32
  const float* alpha  = (const float*)d_in[3];   // ()      f32
  const int*   erow   = (const int*)d_in[4];     // (NNZ,)  int
  const int*   ecol   = (const int*)d_in[5];     // (NNZ,)  int

  const long NH  = in_sizes[0];      // N * 128
  const long nnz = in_sizes[1];
  const int  Kp1 = in_sizes[2];      // 16
  const long n4  = NH / 4;

  char*  ws   = (char*)d_ws;
  float* w    = (float*)ws;                 // softmax weights (16 f32)
  float* bufS = (float*)(ws + 256);         // SpMM accumulator
  float* bufB = bufS + NH;                  // rotating T buffer
  float* bufC = bufB + NH;                  // rotating T buffer
  float* acc  = (float*)d_out;              // output accumulator

  const int  TB = 256;
  const int  GE = 4096;                     // elementwise grid
  const long ntiles = (nnz + TILE - 1) / TILE;
  const int  GS = (int)((ntiles < 2048) ? ntiles : 2048);

  k_softmax<<<1, 32, 0, stream>>>(logits, Kp1, w);

  // S0 = L @ x  (shared by T1 and the high-pass term)
  k_zero<<<GE, TB, 0, stream>>>((float4*)bufS, n4);
  k_spmm<<<GS, TB, 0, stream>>>(vals, erow, ecol, x, bufS, nnz);
  k_init<<<GE, TB, 0, stream>>>((const float4*)x, (const float4*)bufS, w, alpha,
                                (float4*)bufB, (float4*)acc, n4);

  const float* T0p = x;
  float* T1p   = bufB;
  float* spare = bufC;
  for (int k = 2; k < Kp1; ++k) {
    k_zero<<<GE, TB, 0, stream>>>((float4*)bufS, n4);
    k_spmm<<<GS, TB, 0, stream>>>(vals, erow, ecol, T1p, bufS, nnz);
    float* Tn = (T0p == x) ? spare : (float*)T0p;   // recycle dead T0 buffer
    k_step<<<GE, TB, 0, stream>>>((const float4*)bufS, (const float4*)T1p,
                                  (const float4*)T0p, (float4*)Tn,
                                  (float4*)acc, w, k, n4);
    T0p = T1p;
    T1p = Tn;
  }
}